// RelationalUpdate_70978629533888
// MI455X (gfx1250) — compile-verified
//
#include <hip/hip_runtime.h>

// RGCN message computation for MI455X (gfx1250, wave32, WMMA).
//   messages[e, :] = nodes[senders[e], :] (1x64)  @  kernels[edge_types[e]] (64x64)
//
// Bucket edges by relation (16 buckets), then batched GEMM with
// V_WMMA_F32_16X16X4_F32: one wave per 16-edge tile, B (relation kernel)
// staged transposed in LDS (+1 pad, conflict-free b64 reads), A gathered
// per-lane from the node table, C = 4 x v8f accumulators (N = 64).
// k-outer / n-inner ordering keeps 4 independent WMMA accumulation chains.
//
// Roofline: ~34 MB unavoidable traffic @ 23.3 TB/s -> ~1.5 us; 0.54 GFLOP
// needs ~370 TFLOP/s f32 matrix rate to stay memory bound, hence WMMA f32.

typedef __attribute__((ext_vector_type(2))) float v2f;
typedef __attribute__((ext_vector_type(8))) float v8f;

#define N_EDGES 65536
#define D_IN    64
#define D_OUT   64
#define N_REL   16
#define TILE_M  16
#define LDS_STRIDE (D_IN + 1)
// Sum over relations of ceil(count/16) <= E/16 + N_REL
#define MAX_TILES (N_EDGES / TILE_M + N_REL)   // 4112

// d_ws layout (ints):
//   [0..15]   counts[16]
//   [16..32]  tileOff[17]   (exclusive prefix over ceil(count/16))
//   [33..49]  startOff[17]  (exclusive prefix over count)
//   [50..65]  cursors[16]
//   [66..66+N_EDGES) perm   (edge ids grouped by relation)
#define WS_COUNTS   0
#define WS_TILEOFF  16
#define WS_STARTOFF 33
#define WS_CURSORS  50
#define WS_PERM     66

__global__ void rgcn_zero(int* ws) {
    int i = threadIdx.x;
    if (i < WS_PERM) ws[i] = 0;
}

__global__ void rgcn_count(const int* __restrict__ etypes, int* ws) {
    int e = blockIdx.x * blockDim.x + threadIdx.x;
    if (e < N_EDGES) atomicAdd(&ws[WS_COUNTS + etypes[e]], 1);
}

__global__ void rgcn_scan(int* ws) {
    if (threadIdx.x == 0 && blockIdx.x == 0) {
        int s = 0, t = 0;
        for (int r = 0; r < N_REL; ++r) {
            ws[WS_STARTOFF + r] = s;
            ws[WS_TILEOFF + r] = t;
            int c = ws[WS_COUNTS + r];
            s += c;
            t += (c + TILE_M - 1) / TILE_M;
        }
        ws[WS_STARTOFF + N_REL] = s;
        ws[WS_TILEOFF + N_REL] = t;
    }
}

__global__ void rgcn_scatter(const int* __restrict__ etypes, int* ws) {
    int e = blockIdx.x * blockDim.x + threadIdx.x;
    if (e < N_EDGES) {
        int r = etypes[e];
        int pos = ws[WS_STARTOFF + r] + atomicAdd(&ws[WS_CURSORS + r], 1);
        ws[WS_PERM + pos] = e;
    }
}

// One wave (32 lanes) per 16-edge tile. Computes a full 16x64 output slab.
__global__ __launch_bounds__(32)
void rgcn_gemm(const float* __restrict__ nodes,
               const int*   __restrict__ senders,
               const float* __restrict__ kernels,
               const int*   __restrict__ ws,
               float*       __restrict__ out) {
    // Transposed kernel in LDS: ldsW[col*65 + row] = W[row][col]; +1 pad kills
    // the 16-way bank conflict on the column-major reads below.
    __shared__ float ldsW[D_OUT * LDS_STRIDE];
    __shared__ int   sEdge[TILE_M];

    const int t = blockIdx.x;
    if (t >= ws[WS_TILEOFF + N_REL]) return;

    // Locate relation bucket for this flat tile id (uniform across the wave).
    int r = 0;
#pragma unroll
    for (int i = 1; i < N_REL; ++i) r += (t >= ws[WS_TILEOFF + i]) ? 1 : 0;

    const int localTile = t - ws[WS_TILEOFF + r];
    const int segStart  = ws[WS_STARTOFF + r];
    const int cnt       = ws[WS_COUNTS + r];
    const int rowBase   = localTile * TILE_M;
    const int validRows = (cnt - rowBase < TILE_M) ? (cnt - rowBase) : TILE_M;

    const int lane = threadIdx.x;
    const int mLo  = lane & 15;   // row (A) / column (B,C) index within half
    const int half = lane >> 4;   // selects K pair {0,1} vs {2,3}

    // Stage W[r]^T into LDS (128 coalesced global loads/lane, scatter stride 65).
    const float* W = kernels + r * (D_IN * D_OUT);
#pragma unroll 4
    for (int i = 0; i < (D_IN * D_OUT) / 32; ++i) {
        int flat = i * 32 + lane;
        int row = flat >> 6, col = flat & 63;
        ldsW[col * LDS_STRIDE + row] = W[flat];
    }
    // Tile's edge ids (clamped duplicate for ragged tail; stores are masked).
    if (lane < TILE_M) {
        int rr = rowBase + lane;
        sEdge[lane] = ws[WS_PERM + segStart + ((rr < cnt) ? rr : (cnt - 1))];
    }
    __syncthreads();

    // Preload A fragments: 16 K-steps of 16x4 f32.
    // ISA layout: lanes 0-15 hold M=lane, K={4k,4k+1}; lanes 16-31 hold
    // M=lane-16, K={4k+2,4k+3}. Each lane reads one float2 per K-step.
    const float* xrow = nodes + (long)senders[sEdge[mLo]] * D_IN;
    v2f a[16];
#pragma unroll
    for (int k = 0; k < 16; ++k)
        a[k] = *(const v2f*)(xrow + 4 * k + 2 * half);

    // Hoist per-lane output rows out of the masked epilogue:
    // this lane stores C rows m = j + 8*half, j = 0..7, at column mLo (+16n).
    float* pOut[8];
    bool   vOut[8];
#pragma unroll
    for (int j = 0; j < 8; ++j) {
        int m = j + 8 * half;
        vOut[j] = (m < validRows);
        pOut[j] = out + (long)sEdge[m] * D_OUT + mLo;
    }

    // k-outer / n-inner: 4 independent accumulation chains across N-tiles.
    v8f c[4] = {};
#pragma unroll
    for (int k = 0; k < 16; ++k) {
#pragma unroll
        for (int n = 0; n < 4; ++n) {
            // B fragment: column (16n + mLo) of W, rows 4k+2*half .. +1,
            // one 8-byte LDS read (conflict-free via +1 padding).
            v2f b = *(const v2f*)(ldsW + (16 * n + mLo) * LDS_STRIDE
                                       + 4 * k + 2 * half);
            c[n] = __builtin_amdgcn_wmma_f32_16x16x4_f32(
                       /*neg_a=*/false, a[k], /*neg_b=*/false, b,
                       /*c_mod=*/(short)0, c[n],
                       /*reuse_a=*/false, /*reuse_b=*/false);
        }
    }

    // Store: per (j, n) a half-wave writes a 64B contiguous segment.
#pragma unroll
    for (int j = 0; j < 8; ++j) {
        if (vOut[j]) {
#pragma unroll
            for (int n = 0; n < 4; ++n)
                pOut[j][16 * n] = c[n][j];
        }
    }
}

extern "C" void kernel_launch(void* const* d_in, const int* in_sizes, int n_in,
                              void* d_out, int out_size, void* d_ws, size_t ws_size,
                              hipStream_t stream) {
    const float* nodes   = (const float*)d_in[0];
    const int*   senders = (const int*)d_in[1];
    const int*   etypes  = (const int*)d_in[2];
    const float* kerns   = (const float*)d_in[3];
    float*       out     = (float*)d_out;
    int*         ws      = (int*)d_ws;   // needs (66 + N_EDGES)*4 ~ 262 KB

    rgcn_zero   <<<1, 128, 0, stream>>>(ws);
    rgcn_count  <<<N_EDGES / 256, 256, 0, stream>>>(etypes, ws);
    rgcn_scan   <<<1, 32, 0, stream>>>(ws);
    rgcn_scatter<<<N_EDGES / 256, 256, 0, stream>>>(etypes, ws);
    rgcn_gemm   <<<MAX_TILES, 32, 0, stream>>>(nodes, senders, kerns, ws, out);
}